// MultiHeadAttention_21045339750549
// MI455X (gfx1250) — compile-verified
//
#include <hip/hip_runtime.h>
#include <hip/hip_bf16.h>

// Problem constants (from reference): B=2, L=128, D=256, H=8, DK=32, S=4
#define Bb 2
#define Ll 128
#define Dd 256
#define Hh 8
#define DKk 32
#define BL (Bb*Ll)          // 256
#define HDK (Hh*DKk)        // 256

typedef __attribute__((ext_vector_type(2))) float v2f;
typedef __attribute__((ext_vector_type(8))) float v8f;
typedef int vi4 __attribute__((vector_size(16)));   // GCC-style, matches builtin

typedef __attribute__((address_space(1))) vi4* g128p;
typedef __attribute__((address_space(3))) vi4* l128p;
typedef __attribute__((address_space(1))) int* g32p;
typedef __attribute__((address_space(3))) int* l32p;

// ---------------------------------------------------------------------------
// gfx1250 async global->LDS staging helpers (ASYNCcnt path), with portable
// fallbacks so the file compiles on toolchains lacking the builtins.
// ---------------------------------------------------------------------------
__device__ __forceinline__ void cp16_async(const float* g, float* l) {
#if __has_builtin(__builtin_amdgcn_global_load_async_to_lds_b128)
  __builtin_amdgcn_global_load_async_to_lds_b128((g128p)g, (l128p)l, 0, 0);
#else
  l[0] = g[0]; l[1] = g[1]; l[2] = g[2]; l[3] = g[3];
#endif
}

__device__ __forceinline__ void cp4_async(const float* g, float* l) {
#if __has_builtin(__builtin_amdgcn_global_load_async_to_lds_b32)
  __builtin_amdgcn_global_load_async_to_lds_b32((g32p)g, (l32p)l, 0, 0);
#else
  l[0] = g[0];
#endif
}

__device__ __forceinline__ void async_drain() {
#if __has_builtin(__builtin_amdgcn_s_wait_asynccnt)
  __builtin_amdgcn_s_wait_asynccnt(0);
#else
  asm volatile("s_wait_asynccnt 0x0" ::: "memory");
#endif
}

// ---------------------------------------------------------------------------
// Kernel 1: LayerNorm over last dim (256) of q.
// ---------------------------------------------------------------------------
__global__ void ln_kernel(const float* __restrict__ q,
                          const float* __restrict__ gamma,
                          const float* __restrict__ beta,
                          float* __restrict__ qn) {
  int row = blockIdx.x;        // 0..BL-1
  int tid = threadIdx.x;       // 0..255
  __shared__ float red[256];
  float x = q[row * Dd + tid];
  red[tid] = x; __syncthreads();
  for (int s = 128; s > 0; s >>= 1) { if (tid < s) red[tid] += red[tid + s]; __syncthreads(); }
  float mu = red[0] * (1.0f / Dd); __syncthreads();
  float d = x - mu;
  red[tid] = d * d; __syncthreads();
  for (int s = 128; s > 0; s >>= 1) { if (tid < s) red[tid] += red[tid + s]; __syncthreads(); }
  float var = red[0] * (1.0f / Dd);
  float r = rsqrtf(var + 1e-6f);
  qn[row * Dd + tid] = d * r * gamma[tid] + beta[tid];
}

// ---------------------------------------------------------------------------
// Kernel 2: fused projections qp = LN(q)@Wq+bq, kp = k@Wk+bk, vp = v@Wv+bv.
// 256x256 @ 256x256 fp32 GEMM via V_WMMA_F32_16X16X4_F32.
// The 256x16 B-panel is async-staged to LDS once per block (shared by 8 waves).
// grid (16, 2, 3), block (32, 8).
// ---------------------------------------------------------------------------
__global__ void proj_kernel(const float* __restrict__ qn,
                            const float* __restrict__ k,
                            const float* __restrict__ v,
                            const float* __restrict__ Wq, const float* __restrict__ bq,
                            const float* __restrict__ Wk, const float* __restrict__ bk,
                            const float* __restrict__ Wv, const float* __restrict__ bv,
                            float* __restrict__ qp, float* __restrict__ kp,
                            float* __restrict__ vp) {
  const float *X, *W, *bias; float* O;
  int which = blockIdx.z;
  if (which == 0)      { X = qn; W = Wq; bias = bq; O = qp; }
  else if (which == 1) { X = k;  W = Wk; bias = bk; O = kp; }
  else                 { X = v;  W = Wv; bias = bv; O = vp; }
  int lane = threadIdx.x;
  int mr = lane & 15, kh = lane >> 4;
  int tn = blockIdx.x * 16;
  int tm = (blockIdx.y * 8 + threadIdx.y) * 16;

  __shared__ float lds_b[Dd * 16];           // [k][n], 16 KB
  {
    int tid = threadIdx.y * 32 + lane;       // 0..255 -> one K-row each
    const float* g = W + tid * HDK + tn;
    float* l = &lds_b[tid * 16];
    cp16_async(g + 0, l + 0);
    cp16_async(g + 4, l + 4);
    cp16_async(g + 8, l + 8);
    cp16_async(g + 12, l + 12);
    async_drain();
  }
  __syncthreads();

  v8f acc = {};
  for (int k0 = 0; k0 < Dd; k0 += 4) {
    int ka = k0 + 2 * kh;
    v2f a, bf;
    a.x = X[(tm + mr) * Dd + ka];
    a.y = X[(tm + mr) * Dd + ka + 1];
    bf.x = lds_b[ka * 16 + mr];
    bf.y = lds_b[(ka + 1) * 16 + mr];
    acc = __builtin_amdgcn_wmma_f32_16x16x4_f32(false, a, false, bf, (short)0, acc,
                                                false, false);
  }
  float bc = bias[tn + mr];
  int rb = tm + 8 * kh;
  for (int r = 0; r < 8; ++r)
    O[(rb + r) * HDK + tn + mr] = acc[r] + bc;
}

// ---------------------------------------------------------------------------
// Kernel 3: per-(b,h) Gram matrices  G[i][j] = <q_h(i), k_h(j)>, 128x128 K=32.
// The 16-col K^T panel (32x16) is async-staged to LDS.
// grid (8, 1, 16), block (32, 8).
// ---------------------------------------------------------------------------
__global__ void gram_kernel(const float* __restrict__ qp,
                            const float* __restrict__ kp,
                            float* __restrict__ gram) {
  int bh = blockIdx.z; int b = bh >> 3; int h = bh & 7;
  int lane = threadIdx.x; int mr = lane & 15; int kh = lane >> 4;
  int tm = threadIdx.y * 16; int tn = blockIdx.x * 16;
  const float* Qb = qp + b * Ll * HDK + h * DKk;
  const float* Kb = kp + b * Ll * HDK + h * DKk;

  __shared__ float lds_k[DKk * 16];          // [k][n] = K_h[n][k], 2 KB
  {
    int tid = threadIdx.y * 32 + lane;       // 0..255
    int n = tid >> 4, k0 = tid & 15;
    cp4_async(Kb + n * HDK + k0,      &lds_k[k0 * 16 + n]);
    cp4_async(Kb + n * HDK + k0 + 16, &lds_k[(k0 + 16) * 16 + n]);
    async_drain();
  }
  __syncthreads();

  v8f acc = {};
  for (int k0 = 0; k0 < DKk; k0 += 4) {
    int ka = k0 + 2 * kh;
    v2f a, bf;
    a.x = Qb[(tm + mr) * HDK + ka];
    a.y = Qb[(tm + mr) * HDK + ka + 1];
    bf.x = lds_k[ka * 16 + mr];
    bf.y = lds_k[(ka + 1) * 16 + mr];
    acc = __builtin_amdgcn_wmma_f32_16x16x4_f32(false, a, false, bf, (short)0, acc,
                                                false, false);
  }
  float* G = gram + bh * Ll * Ll;
  int rb = tm + 8 * kh;
  for (int r = 0; r < 8; ++r)
    G[(rb + r) * Ll + tn + mr] = acc[r];
}

// ---------------------------------------------------------------------------
// Kernel 4: closed-form logits + softmax + attn*P row sums.
// logit = (A*G_ii + Bc*(G_ij+G_ji) + C*G_jj)/sqrt(DK),
// u = 1+0.1(t_i-t_j), C=(14/9)u^2, Bc=2u-C, A=4-4u+C.
// grid (B*H*L)=2048 blocks of 128 threads (one per j).
// ---------------------------------------------------------------------------
__global__ void softmax_kernel(const float* __restrict__ gram,
                               const float* __restrict__ t,
                               float* __restrict__ attn,
                               float* __restrict__ rowP) {
  int idx = blockIdx.x;           // bh*128 + i
  int bh = idx >> 7; int i = idx & 127; int b = bh >> 3;
  int j = threadIdx.x;
  const float* G = gram + bh * Ll * Ll;
  float u = 1.0f + 0.1f * (t[b * Ll + i] - t[b * Ll + j]);
  float C = (14.0f / 9.0f) * u * u;
  float Bc = 2.0f * u - C;
  float A = 4.0f - 4.0f * u + C;
  float logit = (A * G[i * Ll + i] + Bc * (G[i * Ll + j] + G[j * Ll + i]) +
                 C * G[j * Ll + j]) * 0.17677669529663687f;  // 1/sqrt(32)
  __shared__ float red[128];
  red[j] = logit; __syncthreads();
  for (int s = 64; s > 0; s >>= 1) { if (j < s) red[j] = fmaxf(red[j], red[j + s]); __syncthreads(); }
  float mx = red[0]; __syncthreads();
  float e = __expf(logit - mx);
  red[j] = e; __syncthreads();
  for (int s = 64; s > 0; s >>= 1) { if (j < s) red[j] += red[j + s]; __syncthreads(); }
  float inv = 1.0f / red[0]; __syncthreads();
  float a_ij = e * inv;
  attn[idx * Ll + j] = a_ij;
  red[j] = a_ij * (4.0f - 2.0f * u);   // P weight
  __syncthreads();
  for (int s = 64; s > 0; s >>= 1) { if (j < s) red[j] += red[j + s]; __syncthreads(); }
  if (j == 0) rowP[idx] = red[0];
}

// ---------------------------------------------------------------------------
// Kernel 5: head output = rowP*v_i + (attn .* Q) @ V_h  (128x32, K=128 GEMM).
// The 128x16 V_h panel is async-staged to LDS.
// grid (2, 1, 16), block (32, 8).
// ---------------------------------------------------------------------------
__global__ void attnv_kernel(const float* __restrict__ attn,
                             const float* __restrict__ t,
                             const float* __restrict__ vp,
                             const float* __restrict__ rowP,
                             float* __restrict__ headout) {
  int bh = blockIdx.z; int b = bh >> 3; int h = bh & 7;
  int lane = threadIdx.x; int nr = lane & 15; int kh = lane >> 4;
  int tm = threadIdx.y * 16; int tn = blockIdx.x * 16;
  const float* Ab = attn + bh * Ll * Ll;
  const float* tb = t + b * Ll;
  const float* Vb = vp + b * Ll * HDK + h * DKk;

  __shared__ float lds_v[Ll * 16];           // [k][n] = V_h[k][tn+n], 8 KB
  {
    int tid = threadIdx.y * 32 + lane;       // 0..255
    int kk = tid >> 1, nch = (tid & 1) * 8;
    const float* g = Vb + kk * HDK + tn + nch;
    float* l = &lds_v[kk * 16 + nch];
    cp16_async(g + 0, l + 0);
    cp16_async(g + 4, l + 4);
    async_drain();
  }
  __syncthreads();

  int mi = tm + nr;                 // A row index for this lane
  float ti = tb[mi];
  v8f acc = {};
  for (int k0 = 0; k0 < Ll; k0 += 4) {
    int ka = k0 + 2 * kh;
    v2f a, bf;
    float q0 = 2.0f * (1.0f + 0.1f * (ti - tb[ka]));
    float q1 = 2.0f * (1.0f + 0.1f * (ti - tb[ka + 1]));
    a.x = Ab[mi * Ll + ka] * q0;
    a.y = Ab[mi * Ll + ka + 1] * q1;
    bf.x = lds_v[ka * 16 + nr];
    bf.y = lds_v[(ka + 1) * 16 + nr];
    acc = __builtin_amdgcn_wmma_f32_16x16x4_f32(false, a, false, bf, (short)0, acc,
                                                false, false);
  }
  int rb = tm + 8 * kh;
  for (int r = 0; r < 8; ++r) {
    int i = rb + r;
    headout[(b * Ll + i) * HDK + h * DKk + tn + nr] =
        acc[r] + rowP[bh * Ll + i] * lds_v[i * 16 + nr];
  }
}

// ---------------------------------------------------------------------------
// Kernel 6: out = headout @ fcW + fcb + residual(q).   256x256x256 GEMM.
// B-panel async-staged to LDS.  grid (16, 2, 1), block (32, 8).
// ---------------------------------------------------------------------------
__global__ void fc_kernel(const float* __restrict__ X,
                          const float* __restrict__ W,
                          const float* __restrict__ bias,
                          const float* __restrict__ resid,
                          float* __restrict__ out) {
  int lane = threadIdx.x; int mr = lane & 15; int kh = lane >> 4;
  int tn = blockIdx.x * 16;
  int tm = (blockIdx.y * 8 + threadIdx.y) * 16;

  __shared__ float lds_b[HDK * 16];          // [k][n], 16 KB
  {
    int tid = threadIdx.y * 32 + lane;
    const float* g = W + tid * Dd + tn;
    float* l = &lds_b[tid * 16];
    cp16_async(g + 0, l + 0);
    cp16_async(g + 4, l + 4);
    cp16_async(g + 8, l + 8);
    cp16_async(g + 12, l + 12);
    async_drain();
  }
  __syncthreads();

  v8f acc = {};
  for (int k0 = 0; k0 < HDK; k0 += 4) {
    int ka = k0 + 2 * kh;
    v2f a, bf;
    a.x = X[(tm + mr) * HDK + ka];
    a.y = X[(tm + mr) * HDK + ka + 1];
    bf.x = lds_b[ka * 16 + mr];
    bf.y = lds_b[(ka + 1) * 16 + mr];
    acc = __builtin_amdgcn_wmma_f32_16x16x4_f32(false, a, false, bf, (short)0, acc,
                                                false, false);
  }
  float bc = bias[tn + mr];
  int rb = tm + 8 * kh;
  for (int r = 0; r < 8; ++r) {
    int row = rb + r; int col = tn + mr;
    out[row * Dd + col] = acc[r] + bc + resid[row * Dd + col];
  }
}

// ---------------------------------------------------------------------------
extern "C" void kernel_launch(void* const* d_in, const int* in_sizes, int n_in,
                              void* d_out, int out_size, void* d_ws, size_t ws_size,
                              hipStream_t stream) {
  (void)in_sizes; (void)n_in; (void)out_size; (void)ws_size;
  const float* q     = (const float*)d_in[0];
  const float* k     = (const float*)d_in[1];
  const float* v     = (const float*)d_in[2];
  const float* t     = (const float*)d_in[3];
  const float* Wq    = (const float*)d_in[4];
  const float* bq    = (const float*)d_in[5];
  const float* Wk    = (const float*)d_in[6];
  const float* bk    = (const float*)d_in[7];
  const float* Wv    = (const float*)d_in[8];
  const float* bv    = (const float*)d_in[9];
  const float* fcW   = (const float*)d_in[10];
  const float* fcb   = (const float*)d_in[11];
  const float* gamma = (const float*)d_in[12];
  const float* beta  = (const float*)d_in[13];

  float* ws      = (float*)d_ws;
  float* qn      = ws;                    // 65536
  float* qp      = ws + 65536;            // 65536
  float* kp      = ws + 131072;           // 65536
  float* vp      = ws + 196608;           // 65536
  float* gram    = ws + 262144;           // 262144
  float* rowP    = ws + 524288;           // 2048
  float* headout = ws + 526336;           // 65536

  float* out  = (float*)d_out;            // (B,L,D) = 65536
  float* attn = out + Bb * Ll * Dd;       // (B,H,L,L) = 262144

  dim3 blk(32, 8);
  ln_kernel<<<BL, 256, 0, stream>>>(q, gamma, beta, qn);
  proj_kernel<<<dim3(16, 2, 3), blk, 0, stream>>>(qn, k, v, Wq, bq, Wk, bk, Wv, bv,
                                                  qp, kp, vp);
  gram_kernel<<<dim3(8, 1, 16), blk, 0, stream>>>(qp, kp, gram);
  softmax_kernel<<<Bb * Hh * Ll, 128, 0, stream>>>(gram, t, attn, rowP);
  attnv_kernel<<<dim3(2, 1, 16), blk, 0, stream>>>(attn, t, vp, rowP, headout);
  fc_kernel<<<dim3(16, 2, 1), blk, 0, stream>>>(headout, fcW, fcb, q, out);
}